// Encoder_2963527434889
// MI455X (gfx1250) — compile-verified
//
#include <hip/hip_runtime.h>
#include <hip/hip_bf16.h>

#define NN 100000
#define NE 625000
#define DD 128
#define HH 128
#define LL 2
#define NEG_SLOPE 0.2f
#define BN_EPS 1e-5f

typedef float v2f __attribute__((ext_vector_type(2)));
typedef float v8f __attribute__((ext_vector_type(8)));

// ---------- helpers ----------
__device__ __forceinline__ float leaky(float v) {
    return v >= 0.0f ? v : NEG_SLOPE * v;
}
// monotonic float <-> ordered-uint encoding for atomicMax on floats
__device__ __forceinline__ unsigned fenc(float f) {
    unsigned u = __float_as_uint(f);
    return (u & 0x80000000u) ? ~u : (u | 0x80000000u);
}
__device__ __forceinline__ float fdec(unsigned u) {
    return __uint_as_float((u & 0x80000000u) ? (u ^ 0x80000000u) : ~u);
}

// ---------- 1) fused dual GEMM: xs = x@Wl + bl ; xd = x@Wr + br ----------
// grid.x = N/16 row tiles, block = 256 (8 waves); wave w owns column tile w.
// WMMA f32 16x16x4: A 16x4 (v2f/lane), B 4x16 (v2f/lane), C 16x16 (v8f/lane).
__global__ __launch_bounds__(256) void gat_gemm_dual(
    const float* __restrict__ x,
    const float* __restrict__ Wl, const float* __restrict__ bl,
    const float* __restrict__ Wr, const float* __restrict__ br,
    float* __restrict__ xs, float* __restrict__ xd)
{
    const int wave = threadIdx.x >> 5;
    const int lane = threadIdx.x & 31;
    const int row0 = blockIdx.x * 16;
    const int col0 = wave * 16;
    const int lh   = lane & 15;          // row (A) / col (B) within tile
    const int kq   = (lane >> 4) * 2;    // K sub-offset: 0 or 2

    const float2* a2 = (const float2*)(x + (size_t)(row0 + lh) * DD);

    v8f cl = {};
    v8f cr = {};
    for (int k = 0; k < DD; k += 4) {
        // A fragment: x[row0+lh][k+kq], x[row0+lh][k+kq+1]  (contiguous pair)
        float2 av = a2[(k + kq) >> 1];
        v2f a; a.x = av.x; a.y = av.y;
        // B fragment: lane lh holds column col0+lh, K rows k+kq, k+kq+1
        v2f bL, bR;
        bL.x = Wl[(size_t)(k + kq)     * HH + col0 + lh];
        bL.y = Wl[(size_t)(k + kq + 1) * HH + col0 + lh];
        bR.x = Wr[(size_t)(k + kq)     * HH + col0 + lh];
        bR.y = Wr[(size_t)(k + kq + 1) * HH + col0 + lh];
        cl = __builtin_amdgcn_wmma_f32_16x16x4_f32(false, a, false, bL, (short)0, cl, false, false);
        cr = __builtin_amdgcn_wmma_f32_16x16x4_f32(false, a, false, bR, (short)0, cr, false, false);
    }
    // C layout: lane l, VGPR j -> row = j + 8*(l>>4), col = l&15
    const float biasL = bl[col0 + lh];
    const float biasR = br[col0 + lh];
    const int rbase = row0 + (lane >> 4) * 8;
    #pragma unroll
    for (int j = 0; j < 8; ++j) {
        size_t idx = (size_t)(rbase + j) * HH + col0 + lh;
        xs[idx] = cl[j] + biasL;
        xd[idx] = cr[j] + biasR;
    }
}

// ---------- 6) residual GEMM: out2 = out + out@Wlin + blin ----------
__global__ __launch_bounds__(256) void gat_gemm_res(
    const float* __restrict__ out,
    const float* __restrict__ Wlin, const float* __restrict__ blin,
    float* __restrict__ out2)
{
    const int wave = threadIdx.x >> 5;
    const int lane = threadIdx.x & 31;
    const int row0 = blockIdx.x * 16;
    const int col0 = wave * 16;
    const int lh   = lane & 15;
    const int kq   = (lane >> 4) * 2;

    const float2* a2 = (const float2*)(out + (size_t)(row0 + lh) * HH);

    v8f c = {};
    for (int k = 0; k < HH; k += 4) {
        float2 av = a2[(k + kq) >> 1];
        v2f a; a.x = av.x; a.y = av.y;
        v2f b;
        b.x = Wlin[(size_t)(k + kq)     * HH + col0 + lh];
        b.y = Wlin[(size_t)(k + kq + 1) * HH + col0 + lh];
        c = __builtin_amdgcn_wmma_f32_16x16x4_f32(false, a, false, b, (short)0, c, false, false);
    }
    const float bias = blin[col0 + lh];
    const int rbase = row0 + (lane >> 4) * 8;
    #pragma unroll
    for (int j = 0; j < 8; ++j) {
        size_t idx = (size_t)(rbase + j) * HH + col0 + lh;
        out2[idx] = out[idx] + c[j] + bias;   // residual + bias
    }
}

// ---------- 2) edge logits + fused segment max ----------
// one wave per edge; lanes cover H=128 via float4
__global__ __launch_bounds__(256) void gat_edge_logits(
    const float* __restrict__ xs, const float* __restrict__ xd,
    const int* __restrict__ src, const int* __restrict__ dst,
    const float* __restrict__ att,
    float* __restrict__ logits, unsigned* __restrict__ menc)
{
    const int e = blockIdx.x * (blockDim.x >> 5) + (threadIdx.x >> 5);
    if (e >= NE) return;
    const int lane = threadIdx.x & 31;
    const int s = src[e], d = dst[e];
    const float4 vs = ((const float4*)(xs + (size_t)s * HH))[lane];
    const float4 vd = ((const float4*)(xd + (size_t)d * HH))[lane];
    const float4 a4 = ((const float4*)att)[lane];
    float p = leaky(vs.x + vd.x) * a4.x
            + leaky(vs.y + vd.y) * a4.y
            + leaky(vs.z + vd.z) * a4.z
            + leaky(vs.w + vd.w) * a4.w;
    #pragma unroll
    for (int off = 16; off > 0; off >>= 1) p += __shfl_xor(p, off, 32);
    if (lane == 0) {
        logits[e] = p;
        atomicMax(menc + d, fenc(p));
    }
}

// ---------- 3) a = exp(logit - m[dst]); denom += a (overwrites logits) ----------
__global__ void gat_edge_exp(
    float* __restrict__ logits, const int* __restrict__ dst,
    const unsigned* __restrict__ menc, float* __restrict__ denom)
{
    const int e = blockIdx.x * blockDim.x + threadIdx.x;
    if (e >= NE) return;
    const int d = dst[e];
    const float a = expf(logits[e] - fdec(menc[d]));
    logits[e] = a;
    atomicAdd(denom + d, a);
}

// ---------- 4) out[n][c] = cb[c] ----------
__global__ void gat_out_init(float* __restrict__ out, const float* __restrict__ cb)
{
    const size_t i = (size_t)blockIdx.x * blockDim.x + threadIdx.x;
    out[i] = cb[i & (HH - 1)];
}

// ---------- per-layer zero/init of m, denom, stats ----------
__global__ void gat_seg_init(unsigned* __restrict__ menc, float* __restrict__ denom,
                             float* __restrict__ stats)
{
    const int i = blockIdx.x * blockDim.x + threadIdx.x;
    if (i < NN) { menc[i] = 0u; denom[i] = 0.0f; }
    if (i < 2 * HH) stats[i] = 0.0f;
}

// ---------- 5) out[dst] += alpha * xs[src] ----------
__global__ __launch_bounds__(256) void gat_edge_scatter(
    const float* __restrict__ xs, const float* __restrict__ aexp,
    const float* __restrict__ denom,
    const int* __restrict__ src, const int* __restrict__ dst,
    float* __restrict__ out)
{
    const int e = blockIdx.x * (blockDim.x >> 5) + (threadIdx.x >> 5);
    if (e >= NE) return;
    const int lane = threadIdx.x & 31;
    const int s = src[e], d = dst[e];
    const float alpha = aexp[e] / (denom[d] + 1e-16f);
    const float4 vs = ((const float4*)(xs + (size_t)s * HH))[lane];
    float* po = out + (size_t)d * HH + lane * 4;
    atomicAdd(po + 0, alpha * vs.x);
    atomicAdd(po + 1, alpha * vs.y);
    atomicAdd(po + 2, alpha * vs.z);
    atomicAdd(po + 3, alpha * vs.w);
}

// ---------- 7) batchnorm stats: stats[c]=sum, stats[128+c]=sumsq ----------
__global__ __launch_bounds__(256) void gat_bn_stats(
    const float* __restrict__ out2, float* __restrict__ stats)
{
    const int c = threadIdx.x & (HH - 1);
    const int rpb = blockDim.x >> 7;                       // rows per block step
    int r = blockIdx.x * rpb + (threadIdx.x >> 7);
    const int rstride = gridDim.x * rpb;
    float s = 0.0f, q = 0.0f;
    for (; r < NN; r += rstride) {
        const float v = out2[(size_t)r * HH + c];
        s += v; q += v * v;
    }
    atomicAdd(stats + c, s);
    atomicAdd(stats + HH + c, q);
}

// ---------- 8) batchnorm apply (+ optional relu) ----------
__global__ void gat_bn_apply(
    const float* __restrict__ out2, const float* __restrict__ stats,
    const float* __restrict__ gamma, const float* __restrict__ beta,
    float* __restrict__ xnext, int do_relu)
{
    const size_t i = (size_t)blockIdx.x * blockDim.x + threadIdx.x;
    const int c = (int)(i & (HH - 1));
    const float invN = 1.0f / (float)NN;
    const float mu  = stats[c] * invN;
    const float var = stats[HH + c] * invN - mu * mu;
    float v = (out2[i] - mu) * rsqrtf(var + BN_EPS) * gamma[c] + beta[c];
    if (do_relu) v = fmaxf(v, 0.0f);
    xnext[i] = v;
}

extern "C" void kernel_launch(void* const* d_in, const int* in_sizes, int n_in,
                              void* d_out, int out_size, void* d_ws, size_t ws_size,
                              hipStream_t stream) {
    (void)in_sizes; (void)n_in; (void)out_size; (void)ws_size;

    const float* x0    = (const float*)d_in[0];
    const int*   ei    = (const int*)  d_in[1];
    const float* Wl    = (const float*)d_in[2];
    const float* bl    = (const float*)d_in[3];
    const float* Wr    = (const float*)d_in[4];
    const float* br    = (const float*)d_in[5];
    const float* att   = (const float*)d_in[6];
    const float* cb    = (const float*)d_in[7];
    const float* Wlin  = (const float*)d_in[8];
    const float* blin  = (const float*)d_in[9];
    const float* gamma = (const float*)d_in[10];
    const float* beta  = (const float*)d_in[11];

    const int* src = ei;        // edge_index[0]
    const int* dst = ei + NE;   // edge_index[1]

    // workspace layout (floats)
    const size_t NH = (size_t)NN * HH;
    float*    bufA   = (float*)d_ws;        // xs
    float*    bufB   = bufA + NH;           // xd, then reused as `out`
    float*    bufC   = bufB + NH;           // out2
    float*    bufD   = bufC + NH;           // x between layers
    float*    logits = bufD + NH;           // [NE], reused as a=exp(...)
    float*    denom  = logits + NE;         // [NN]
    unsigned* menc   = (unsigned*)(denom + NN); // [NN]
    float*    stats  = (float*)(menc + NN); // [2*HH]

    const dim3 blk(256);
    const dim3 gGemm(NN / 16);                 // 6250 row tiles
    const dim3 gEdgeW((NE + 7) / 8);           // wave per edge, 8 waves/block
    const dim3 gEdgeT((NE + 255) / 256);
    const dim3 gElem((unsigned)(NH / 256));    // 50000
    const dim3 gInit((NN + 255) / 256);
    const dim3 gStats(512);

    for (int i = 0; i < LL; ++i) {
        const float* x = (i == 0) ? x0 : bufD;
        float* xnext   = (i < LL - 1) ? bufD : (float*)d_out;

        gat_seg_init<<<gInit, blk, 0, stream>>>(menc, denom, stats);

        gat_gemm_dual<<<gGemm, blk, 0, stream>>>(
            x, Wl + (size_t)i * DD * HH, bl + (size_t)i * HH,
               Wr + (size_t)i * DD * HH, br + (size_t)i * HH,
            bufA, bufB);

        gat_edge_logits<<<gEdgeW, blk, 0, stream>>>(
            bufA, bufB, src, dst, att + (size_t)i * HH, logits, menc);

        gat_edge_exp<<<gEdgeT, blk, 0, stream>>>(logits, dst, menc, denom);

        // bufB (xd) is dead now -> becomes `out`
        gat_out_init<<<gElem, blk, 0, stream>>>(bufB, cb + (size_t)i * HH);

        gat_edge_scatter<<<gEdgeW, blk, 0, stream>>>(
            bufA, logits, denom, src, dst, bufB);

        gat_gemm_res<<<gGemm, blk, 0, stream>>>(
            bufB, Wlin + (size_t)i * HH * HH, blin + (size_t)i * HH, bufC);

        gat_bn_stats<<<gStats, blk, 0, stream>>>(bufC, stats);

        gat_bn_apply<<<gElem, blk, 0, stream>>>(
            bufC, stats, gamma + (size_t)i * HH, beta + (size_t)i * HH,
            xnext, (i < LL - 1) ? 1 : 0);
    }
}